// SegmentationHistModel_12360915878601
// MI455X (gfx1250) — compile-verified
//
#include <hip/hip_runtime.h>
#include <hip/hip_bf16.h>

// Segmentation confusion-matrix histogram for MI455X (gfx1250, wave32).
// Memory-bound: 134 MB streamed -> ~5.8us floor at 23.3 TB/s.
// Strategy: coalesced b128 loads, per-wave LDS sub-histograms (ds_add_u32),
// float global-atomic flush.

#define BDIM    256
#define WAVES   8          // 256 threads / wave32
#define HPAD    384        // >= 19*19 bins (nc <= 19), padded
#define CHUNKS  32         // chunks per image
#define IMG_HW  (512 * 512)

__global__ void seg_hist_zero(float* __restrict__ out, int n) {
    int i = blockIdx.x * blockDim.x + threadIdx.x;
    if (i < n) out[i] = 0.0f;
}

__global__ __launch_bounds__(BDIM) void seg_hist_kernel(
    const int*   __restrict__ pred,
    const float* __restrict__ segmaps,
    const int*   __restrict__ ncls_p,
    const int*   __restrict__ udc_p,
    float*       __restrict__ out)
{
    __shared__ unsigned int hist[WAVES][HPAD];

    const int nc     = ncls_p[0];
    const int nc2    = nc * nc;
    const int ignore = udc_p[0] ? nc : -1;

    const int tid  = threadIdx.x;
    const int wave = tid >> 5;

    // Zero the privatized LDS histograms.
    unsigned int* flat = &hist[0][0];
    for (int i = tid; i < WAVES * HPAD; i += BDIM) flat[i] = 0u;
    __syncthreads();

    const int img      = blockIdx.x / CHUNKS;
    const int chunk    = blockIdx.x % CHUNKS;
    const int chunkPix = IMG_HW / CHUNKS;                  // 8192
    const long base    = (long)img * IMG_HW + (long)chunk * chunkPix;

    const int4*   pred4 = (const int4*)(pred + base);
    const float4* seg4  = (const float4*)(segmaps + base);
    const int iters = chunkPix / (BDIM * 4);               // 8

    unsigned int (* __restrict__ myhist)[HPAD] = &hist[wave];

    for (int it = 0; it < iters; ++it) {
        const int idx = it * BDIM + tid;                   // coalesced b128
        const int4   p = pred4[idx];
        const float4 s = seg4[idx];

        // gt = int(segmap * 255.0f)  -- exact mirror of reference math
        const int g0 = (int)(s.x * 255.0f);
        const int g1 = (int)(s.y * 255.0f);
        const int g2 = (int)(s.z * 255.0f);
        const int g3 = (int)(s.w * 255.0f);

        if (g0 != ignore) atomicAdd(&(*myhist)[p.x * nc + g0], 1u);
        if (g1 != ignore) atomicAdd(&(*myhist)[p.y * nc + g1], 1u);
        if (g2 != ignore) atomicAdd(&(*myhist)[p.z * nc + g2], 1u);
        if (g3 != ignore) atomicAdd(&(*myhist)[p.w * nc + g3], 1u);
    }
    __syncthreads();

    // Reduce the 8 per-wave sub-histograms, flush with one float atomic
    // per non-empty bin per block. Counts < 2^24 are exact in f32.
    float* imgOut = out + (long)img * nc2;
    for (int bin = tid; bin < nc2; bin += BDIM) {
        unsigned int sum = 0;
        #pragma unroll
        for (int w = 0; w < WAVES; ++w) sum += hist[w][bin];
        if (sum) atomicAdd(&imgOut[bin], (float)sum);
    }
}

extern "C" void kernel_launch(void* const* d_in, const int* in_sizes, int n_in,
                              void* d_out, int out_size, void* d_ws, size_t ws_size,
                              hipStream_t stream) {
    const int*   pred    = (const int*)d_in[0];
    const float* segmaps = (const float*)d_in[1];
    const int*   ncls    = (const int*)d_in[2];
    const int*   udc     = (const int*)d_in[3];
    float*       out     = (float*)d_out;

    const int B = in_sizes[0] / IMG_HW;                    // 64

    // Output is poisoned; zero it first (same stream => ordered).
    seg_hist_zero<<<(out_size + BDIM - 1) / BDIM, BDIM, 0, stream>>>(out, out_size);

    seg_hist_kernel<<<B * CHUNKS, BDIM, 0, stream>>>(pred, segmaps, ncls, udc, out);
}